// PrototypeContrastLoss_83141976916016
// MI455X (gfx1250) — compile-verified
//
#include <hip/hip_runtime.h>

// PrototypeContrastLoss for MI455X (gfx1250, wave32).
//
// Roofline: 304 MB of single-use streaming reads (we skip the unused
// cross_prototype input) vs ~143 MFLOP -> purely HBM-bound (~13 us at
// 23.3 TB/s). Stage 1 is a coalesced b128 non-temporal streaming kernel,
// one wave32 per row. Stage 2 sums the 8192 per-row losses with chained
// V_WMMA_F32_16X16X4_F32 against an all-ones B matrix (deterministic,
// no float atomics).

typedef __attribute__((ext_vector_type(2))) float v2f;
typedef __attribute__((ext_vector_type(4))) float v4f;
typedef __attribute__((ext_vector_type(8))) float v8f;

#define PCL_BS 8192
#define PCL_D  512
#define PCL_K  8
#define PCL_EPS 1e-8f

__device__ __forceinline__ float pcl_dot4(v4f a, v4f b, float acc) {
  acc = fmaf(a.x, b.x, acc);
  acc = fmaf(a.y, b.y, acc);
  acc = fmaf(a.z, b.z, acc);
  acc = fmaf(a.w, b.w, acc);
  return acc;
}

__device__ __forceinline__ float pcl_wave_sum(float x) {
#pragma unroll
  for (int off = 16; off > 0; off >>= 1) x += __shfl_xor(x, off, 32);
  return x;
}

// Stage 1: one wave32 per row. Lane l owns float4 slots {l, l+32, l+64, l+96}
// of the 512-float row -> each 32-lane access is a contiguous 512B segment.
__global__ __launch_bounds__(256) void pcl_row_loss(
    const float* __restrict__ base, const float* __restrict__ pos,
    const float* __restrict__ crossp, const float* __restrict__ neg1,
    const float* __restrict__ neg2, float* __restrict__ row_loss) {
  const int lid = threadIdx.x & 31;
  const int row = blockIdx.x * 8 + (threadIdx.x >> 5);
  const long rb = (long)row * PCL_D;

  // base row (kept resident; reused against 4 other vectors)
  v4f b[4];
  {
    const v4f* p = (const v4f*)(base + rb);
#pragma unroll
    for (int j = 0; j < 4; ++j)
      b[j] = __builtin_nontemporal_load(p + lid + 32 * j);
  }
  float bb = 0.f;
#pragma unroll
  for (int j = 0; j < 4; ++j) bb = pcl_dot4(b[j], b[j], bb);

  float bp, pp, bc, cc;
  {
    v4f s[4];
    // sum_k pos_pro[k] (mean's 1/8 applied at the cosine, scale-faithfully)
    const v4f* p0 = (const v4f*)(pos + rb);
#pragma unroll
    for (int j = 0; j < 4; ++j)
      s[j] = __builtin_nontemporal_load(p0 + lid + 32 * j);
#pragma unroll
    for (int k = 1; k < PCL_K; ++k) {
      const v4f* p = (const v4f*)(pos + (long)k * PCL_BS * PCL_D + rb);
#pragma unroll
      for (int j = 0; j < 4; ++j)
        s[j] += __builtin_nontemporal_load(p + lid + 32 * j);
    }
    bp = 0.f; pp = 0.f;
#pragma unroll
    for (int j = 0; j < 4; ++j) {
      bp = pcl_dot4(b[j], s[j], bp);
      pp = pcl_dot4(s[j], s[j], pp);
    }

    // sum_k cross_p[k] (reuse the same registers)
    const v4f* c0 = (const v4f*)(crossp + rb);
#pragma unroll
    for (int j = 0; j < 4; ++j)
      s[j] = __builtin_nontemporal_load(c0 + lid + 32 * j);
#pragma unroll
    for (int k = 1; k < PCL_K; ++k) {
      const v4f* p = (const v4f*)(crossp + (long)k * PCL_BS * PCL_D + rb);
#pragma unroll
      for (int j = 0; j < 4; ++j)
        s[j] += __builtin_nontemporal_load(p + lid + 32 * j);
    }
    bc = 0.f; cc = 0.f;
#pragma unroll
    for (int j = 0; j < 4; ++j) {
      bc = pcl_dot4(b[j], s[j], bc);
      cc = pcl_dot4(s[j], s[j], cc);
    }
  }

  float bn1 = 0.f, n11 = 0.f, bn2 = 0.f, n22 = 0.f;
  {
    const v4f* p = (const v4f*)(neg1 + rb);
#pragma unroll
    for (int j = 0; j < 4; ++j) {
      v4f t = __builtin_nontemporal_load(p + lid + 32 * j);
      bn1 = pcl_dot4(b[j], t, bn1);
      n11 = pcl_dot4(t, t, n11);
    }
    const v4f* q = (const v4f*)(neg2 + rb);
#pragma unroll
    for (int j = 0; j < 4; ++j) {
      v4f t = __builtin_nontemporal_load(q + lid + 32 * j);
      bn2 = pcl_dot4(b[j], t, bn2);
      n22 = pcl_dot4(t, t, n22);
    }
  }

  // 9 cross-lane reductions, 5 shuffle steps each (wave32)
  bb  = pcl_wave_sum(bb);
  bp  = pcl_wave_sum(bp);  pp  = pcl_wave_sum(pp);
  bc  = pcl_wave_sum(bc);  cc  = pcl_wave_sum(cc);
  bn1 = pcl_wave_sum(bn1); n11 = pcl_wave_sum(n11);
  bn2 = pcl_wave_sum(bn2); n22 = pcl_wave_sum(n22);

  if (lid == 0) {
    const float nb = sqrtf(bb);
    // cosine with the mean's 1/8 folded in on both num and den (faithful
    // to jnp.mean + the eps clamp on the denominator)
    const float c1  = (bp * 0.125f) / fmaxf(nb * (sqrtf(pp) * 0.125f), PCL_EPS);
    const float c3  = (bc * 0.125f) / fmaxf(nb * (sqrtf(cc) * 0.125f), PCL_EPS);
    const float cn1 = bn1 / fmaxf(nb * sqrtf(n11), PCL_EPS);
    const float cn2 = bn2 / fmaxf(nb * sqrtf(n22), PCL_EPS);
    const float posl = expf(c1) + expf(c3);
    const float negl = expf(cn1) + expf(cn2) + posl;
    row_loss[row] = -logf(posl / negl + PCL_EPS);
  }
}

// Stage 2: deterministic sum of 8192 floats via chained f32 WMMA.
// 8 waves; wave w views row_loss[w*1024 .. w*1024+1023] as a 16x64 matrix L.
// Each of 16 chunks supplies A = L[:, 4t:4t+4] in the ISA 16x4 f32 layout
// (lanes 0-15 hold K=0,1; lanes 16-31 hold K=2,3); B = ones(4x16), so
// D[m][n] accumulates the row sums of L. Column 0 of D (lanes 0 & 16) then
// holds rows 0-7 / 8-15 sums in the 8 accumulator VGPRs.
__global__ __launch_bounds__(256) void pcl_reduce_wmma(
    const float* __restrict__ row_loss, float* __restrict__ out) {
  __shared__ float part[8];
  const int lid  = threadIdx.x & 31;
  const int wv   = threadIdx.x >> 5;
  const int m    = lid & 15;
  const int koff = (lid >> 4) * 2;

  const float* ap = row_loss + wv * 1024 + m * 64 + koff;
  v8f c = {};
  const v2f ones = {1.0f, 1.0f};
#pragma unroll
  for (int t = 0; t < 16; ++t) {
    v2f a = *(const v2f*)(ap + t * 4);
    c = __builtin_amdgcn_wmma_f32_16x16x4_f32(
        /*neg_a=*/false, a, /*neg_b=*/false, ones,
        /*c_mod=*/(short)0, c, /*reuse_a=*/false, /*reuse_b=*/false);
  }
  float s = c[0] + c[1] + c[2] + c[3] + c[4] + c[5] + c[6] + c[7];
  s += __shfl_xor(s, 16, 32);  // combine M=0..7 (lane 0) with M=8..15 (lane 16)
  if (lid == 0) part[wv] = s;
  __syncthreads();
  if (threadIdx.x == 0) {
    float tot = 0.f;
#pragma unroll
    for (int w = 0; w < 8; ++w) tot += part[w];
    out[0] = tot * (1.0f / (float)PCL_BS);
  }
}

extern "C" void kernel_launch(void* const* d_in, const int* in_sizes, int n_in,
                              void* d_out, int out_size, void* d_ws, size_t ws_size,
                              hipStream_t stream) {
  (void)in_sizes; (void)n_in; (void)out_size; (void)ws_size;
  const float* base   = (const float*)d_in[0];  // [8192, 512]
  const float* pos    = (const float*)d_in[1];  // [8, 8192, 512]
  // d_in[2] (cross_prototype) feeds only the unused pos_pro_2 -> never read.
  const float* crossp = (const float*)d_in[3];  // [8, 8192, 512]
  const float* neg1   = (const float*)d_in[4];  // [8192, 512]
  const float* neg2   = (const float*)d_in[5];  // [8192, 512]
  float* out = (float*)d_out;                   // shape (1,)
  float* rl  = (float*)d_ws;                    // 8192 floats of scratch

  pcl_row_loss<<<PCL_BS / 8, 256, 0, stream>>>(base, pos, crossp, neg1, neg2, rl);
  pcl_reduce_wmma<<<1, 256, 0, stream>>>(rl, out);
}